// RestrictedDet_32469952758280
// MI455X (gfx1250) — compile-verified
//
#include <hip/hip_runtime.h>

// RestrictedDet slogdet on MI455X (gfx1250, wave32).
// One single-wave workgroup per 64x64 determinant (8192 total), matrix in LDS.
// Blocked LU (block=16) with partial pivoting; trailing rank-16 updates via
// V_WMMA_F32_16X16X4_F32 (fp32 WMMA, chained K=4 x 4).
// L multipliers are stored NEGATED so the WMMA trailing update
// A22 = (-L21)*U12 + A22 needs no per-fragment sign flips (f32 WMMA has no
// A-neg modifier per the ISA NEG table).

#define NSITES 128
#define NUP    64
#define BATCH  4096
#define LDP    65   // LDS row stride (floats); 65 kills 64-bank conflicts

typedef __attribute__((ext_vector_type(2))) float v2f;
typedef __attribute__((ext_vector_type(8))) float v8f;

__global__ __launch_bounds__(32)
void lu_slogdet_kernel(const float* __restrict__ U,
                       const int*   __restrict__ idx_up,
                       const int*   __restrict__ idx_dn,
                       float* __restrict__ ws_sign,
                       float* __restrict__ ws_log)
{
    __shared__ float As[NUP * LDP];

    const int lane = threadIdx.x;        // 0..31 (wave32)
    const int gid  = blockIdx.x;         // 0..8191
    const int sel  = gid >> 12;          // 0 = up, 1 = dn
    const int b    = gid & (BATCH - 1);
    const int* idx = (sel == 0 ? idx_up : idx_dn) + b * NUP;

    // ---- gather: As[r][c] = U[idx[r]][c] (row-coalesced 128B bursts) ----
    for (int i = lane; i < NUP * NUP; i += 32) {
        int r = i >> 6;
        int c = i & 63;
        As[r * LDP + c] = U[idx[r] * NUP + c];
    }
    __syncthreads();

    float sign   = 1.0f;
    float logabs = 0.0f;

    const int half = lane >> 4;   // 0: lanes 0-15, 1: lanes 16-31
    const int l16  = lane & 15;

    for (int kb = 0; kb < NUP; kb += 16) {
        const int pend = kb + 15;

        // ================= panel factorization (cols kb..kb+15) =================
        for (int j = kb; j <= pend; ++j) {
            __syncthreads();
            // --- partial-pivot search in column j, rows j..63 ---
            float best = -1.0f; int bestr = j;
            for (int r = j + lane; r < NUP; r += 32) {
                float av = fabsf(As[r * LDP + j]);
                if (av > best) { best = av; bestr = r; }
            }
            for (int off = 16; off > 0; off >>= 1) {
                float ov  = __shfl_xor(best,  off, 32);
                int   orr = __shfl_xor(bestr, off, 32);
                if (ov > best || (ov == best && orr < bestr)) { best = ov; bestr = orr; }
            }
            if (bestr != j) {            // uniform branch; swap full rows
                for (int c = lane; c < NUP; c += 32) {
                    float t = As[j * LDP + c];
                    As[j * LDP + c]     = As[bestr * LDP + c];
                    As[bestr * LDP + c] = t;
                }
                sign = -sign;
            }
            __syncthreads();
            float diag = As[j * LDP + j];
            sign   *= (diag > 0.f) ? 1.f : ((diag < 0.f) ? -1.f : 0.f);
            logabs += logf(fabsf(diag));           // -inf if singular (sign -> 0)
            float ninvd = -1.0f / diag;
            // negated multipliers + rank-1 update restricted to the panel
            for (int r = j + 1 + lane; r < NUP; r += 32) {
                float nm = As[r * LDP + j] * ninvd;   // nm = -L(r,j)
                As[r * LDP + j] = nm;
                for (int c = j + 1; c <= pend; ++c)
                    As[r * LDP + c] += nm * As[j * LDP + c];
            }
        }
        __syncthreads();

        if (kb + 16 >= NUP) break;       // last block: no trailing work

        // ============== triangular solve: U12 = L11^{-1} * A12 ==============
        // (L11 strictly-lower entries are stored negated -> use +=)
        for (int c0 = kb + 16; c0 < NUP; c0 += 32) {
            int  c   = c0 + lane;
            bool act = (c < NUP);
            int  cc  = act ? c : (NUP - 1);
            float x[16];
            #pragma unroll
            for (int i = 0; i < 16; ++i) x[i] = As[(kb + i) * LDP + cc];
            #pragma unroll
            for (int i = 1; i < 16; ++i) {
                float xi = x[i];
                for (int m = 0; m < i; ++m)
                    xi += As[(kb + i) * LDP + (kb + m)] * x[m];   // negated L11
                x[i] = xi;
            }
            if (act) {
                #pragma unroll
                for (int i = 0; i < 16; ++i) As[(kb + i) * LDP + cc] = x[i];
            }
        }
        __syncthreads();

        // ====== trailing update A22 = (-L21)*U12 + A22 via fp32 WMMA tiles ======
        // EXEC is all-1s here (uniform control flow, single wave).
        // A-fragments are the stored (already-negated) multipliers: no VALU fixup.
        for (int rt = kb + 16; rt < NUP; rt += 16) {
            for (int ct = kb + 16; ct < NUP; ct += 16) {
                // C/D 16x16 f32 fragment: VGPR v -> row rt+v (+8 for lanes 16-31)
                v8f cfrag;
                #pragma unroll
                for (int vv = 0; vv < 8; ++vv)
                    cfrag[vv] = As[(rt + vv + half * 8) * LDP + (ct + l16)];
                #pragma unroll
                for (int kc = 0; kc < 4; ++kc) {
                    const int k0 = kb + 4 * kc;
                    // A 16x4 f32: lane M = l16; VGPR0 holds K=2*half, VGPR1 K=2*half+1
                    v2f a, bb;
                    a[0] = As[(rt + l16) * LDP + (k0 + 2 * half)];
                    a[1] = As[(rt + l16) * LDP + (k0 + 2 * half + 1)];
                    // B 4x16 f32: lane N = l16; rows dual to A layout
                    bb[0] = As[(k0 + 2 * half)     * LDP + (ct + l16)];
                    bb[1] = As[(k0 + 2 * half + 1) * LDP + (ct + l16)];
                    cfrag = __builtin_amdgcn_wmma_f32_16x16x4_f32(
                        /*neg_a=*/false, a, /*neg_b=*/false, bb,
                        /*c_mod=*/(short)0, cfrag,
                        /*reuse_a=*/false, /*reuse_b=*/false);
                }
                #pragma unroll
                for (int vv = 0; vv < 8; ++vv)
                    As[(rt + vv + half * 8) * LDP + (ct + l16)] = cfrag[vv];
            }
        }
        __syncthreads();
    }

    if (lane == 0) {
        ws_sign[gid] = sign;
        ws_log[gid]  = logabs;
    }
}

__global__ void combine_kernel(const float* __restrict__ ws_sign,
                               const float* __restrict__ ws_log,
                               float* __restrict__ out)
{
    int b = blockIdx.x * blockDim.x + threadIdx.x;
    if (b < BATCH) {
        out[b]         = ws_sign[b] * ws_sign[BATCH + b];   // sign_up * sign_dn
        out[BATCH + b] = ws_log[b]  + ws_log[BATCH + b];    // logabs_up + logabs_dn
    }
}

extern "C" void kernel_launch(void* const* d_in, const int* in_sizes, int n_in,
                              void* d_out, int out_size, void* d_ws, size_t ws_size,
                              hipStream_t stream) {
    const float* U      = (const float*)d_in[0];   // (128, 64) f32
    const int*   idx_up = (const int*)  d_in[1];   // (4096, 64) i32
    const int*   idx_dn = (const int*)  d_in[2];   // (4096, 64) i32
    float* out = (float*)d_out;                    // 8192 floats: signs ‖ logabs

    float* ws_sign = (float*)d_ws;                 // 8192 floats
    float* ws_log  = ws_sign + 2 * BATCH;          // 8192 floats

    // 8192 determinants, one single-wave workgroup each
    lu_slogdet_kernel<<<2 * BATCH, 32, 0, stream>>>(U, idx_up, idx_dn,
                                                    ws_sign, ws_log);
    combine_kernel<<<(BATCH + 255) / 256, 256, 0, stream>>>(ws_sign, ws_log, out);
}